// GLNetIndexConv42_12335146074663
// MI455X (gfx1250) — compile-verified
//
#include <hip/hip_runtime.h>
#include <hip/hip_bf16.h>

typedef __attribute__((ext_vector_type(2))) float v2f;
typedef __attribute__((ext_vector_type(8))) float v8f;

// ---------------------------------------------------------------------------
// Gather-conv as per-position GEMM with V_WMMA_F32_16X16X4_F32.
//   out[b, o, n] = sum_{c,k} in[b, c, idx[k,n]] * w[o, c, k] + bias[o]
// One wave computes OT (=4) 16(b) x 16(o) tiles at position n, sharing the
// scattered A gather across the OT accumulators.
// Grid: (Nin, Cout/(16*OT), B/16), block: 32 (one wave).
// ---------------------------------------------------------------------------
template <int Cin, int Nin, int Cout, int BStride, int OT>
__global__ __launch_bounds__(32) void conv_wmma_kernel(
    const float* __restrict__ in,      // (B, Cin, Nin), b-stride = BStride
    const int*   __restrict__ idx,     // (7, Nin)
    const float* __restrict__ w,       // (Cout, Cin, 7)
    const float* __restrict__ bias,    // (Cout)
    float*       __restrict__ out)     // (B, Cout, Nin)
{
    constexpr int Kdim = Cin * 7;
    const int n    = blockIdx.x;
    const int o0   = blockIdx.y * 16 * OT;
    const int b0   = blockIdx.z * 16;
    const int lane = threadIdx.x;     // 0..31
    const int half = lane >> 4;       // selects K pair {2h, 2h+1}
    const int rc   = lane & 15;       // A row (b) / B col (o) / D col (o)

    // Uniform gather indices for this position (scalar loads).
    int gidx[7];
#pragma unroll
    for (int kk = 0; kk < 7; ++kk) gidx[kk] = idx[kk * Nin + n];

    const float* __restrict__ arow = in + (size_t)(b0 + rc) * BStride;
    const float* __restrict__ brow[OT];
#pragma unroll
    for (int ot = 0; ot < OT; ++ot) {
        brow[ot] = w + (size_t)(o0 + ot * 16 + rc) * Kdim;
        __builtin_prefetch(brow[ot], 0, 1);   // global_prefetch_b8
    }

    v8f acc[OT];
#pragma unroll
    for (int ot = 0; ot < OT; ++ot) acc[ot] = (v8f){};

    for (int kb = 0; kb < Kdim; kb += 4) {
        const int k0 = kb + half * 2;
        const int k1 = k0 + 1;
        // A: gathered input, shared by all OT tiles.
        v2f a;
        {
            const int c0 = k0 / 7, kk0 = k0 % 7;
            const int c1 = k1 / 7, kk1 = k1 % 7;
            a.x = (k0 < Kdim) ? arow[c0 * Nin + gidx[kk0]] : 0.0f;
            a.y = (k1 < Kdim) ? arow[c1 * Nin + gidx[kk1]] : 0.0f;
        }
#pragma unroll
        for (int ot = 0; ot < OT; ++ot) {
            v2f b;
            if constexpr ((Kdim & 3) == 0) {
                b = *(const v2f*)(brow[ot] + k0);      // global_load_b64
            } else {
                b.x = (k0 < Kdim) ? brow[ot][k0] : 0.0f;
                b.y = (k1 < Kdim) ? brow[ot][k1] : 0.0f;
            }
            acc[ot] = __builtin_amdgcn_wmma_f32_16x16x4_f32(
                          false, a, false, b, (short)0, acc[ot], false, false);
        }
    }

#pragma unroll
    for (int ot = 0; ot < OT; ++ot) {
        const int o  = o0 + ot * 16 + rc;
        const float bv = bias[o];
#pragma unroll
        for (int j = 0; j < 8; ++j) {
            const int m = j + 8 * half;   // b row within tile
            out[((size_t)(b0 + m) * Cout + o) * Nin + n] = acc[ot][j] + bv;
        }
    }
}

// ---------------------------------------------------------------------------
// Gather max-pool: out[b,o,m] = max_k conv[b,o,idxp[k,m]]
// ---------------------------------------------------------------------------
__global__ __launch_bounds__(256) void pool_kernel(
    const float* __restrict__ conv,    // (B, Cout, Nin)
    const int*   __restrict__ idxp,    // (7, Nout)
    float*       __restrict__ out,     // (B, Cout, Nout)
    int Nin, int Nout, int total)
{
    int i = blockIdx.x * 256 + threadIdx.x;
    if (i >= total) return;
    const int m  = i % Nout;
    const int bo = i / Nout;
    const float* __restrict__ src = conv + (size_t)bo * Nin;
    float v = -INFINITY;
#pragma unroll
    for (int k = 0; k < 7; ++k)
        v = fmaxf(v, src[idxp[k * Nout + m]]);
    out[i] = v;
}

// ---------------------------------------------------------------------------
// Batch-norm statistics over (batch, spatial) per channel. x: (B, C, N).
// One block per channel; writes fused scale/shift for y = x*scale[c]+shift[c].
// ---------------------------------------------------------------------------
__global__ __launch_bounds__(256) void bn_stats_kernel(
    const float* __restrict__ x, int B, int C, int N,
    const float* __restrict__ gamma, const float* __restrict__ beta,
    float* __restrict__ scale, float* __restrict__ shift)
{
    __shared__ float ssum[256];
    __shared__ float ssq[256];
    const int c = blockIdx.x;
    const int total = B * N;
    float s = 0.f, q = 0.f;
    for (int i = threadIdx.x; i < total; i += 256) {
        const int b = i / N, m = i % N;
        const float v = x[((size_t)b * C + c) * N + m];
        s += v; q += v * v;
    }
    ssum[threadIdx.x] = s; ssq[threadIdx.x] = q;
    __syncthreads();
    for (int st = 128; st > 0; st >>= 1) {
        if (threadIdx.x < st) {
            ssum[threadIdx.x] += ssum[threadIdx.x + st];
            ssq[threadIdx.x]  += ssq[threadIdx.x + st];
        }
        __syncthreads();
    }
    if (threadIdx.x == 0) {
        const float inv  = 1.0f / (float)total;
        const float mean = ssum[0] * inv;
        const float var  = ssq[0] * inv - mean * mean;
        const float sc   = gamma[c] * rsqrtf(var + 1e-5f);
        scale[c] = sc;
        shift[c] = beta[c] - mean * sc;
    }
}

// ---------------------------------------------------------------------------
// BN apply + ReLU, elementwise over (B, C, N); safe in-place.
// ---------------------------------------------------------------------------
__global__ __launch_bounds__(256) void bn_apply_relu_kernel(
    const float* __restrict__ x, float* __restrict__ y,
    const float* __restrict__ scale, const float* __restrict__ shift,
    int C, int N, int total)
{
    int i = blockIdx.x * 256 + threadIdx.x;
    if (i >= total) return;
    const int c = (i / N) % C;
    const float v = x[i] * scale[c] + shift[c];
    y[i] = v > 0.f ? v : 0.f;
}

// ---------------------------------------------------------------------------
// lin1 GEMM with V_WMMA_F32_16X16X4_F32; fuses the mean over the two groups.
//   out[b, h] = (0.5*(a0[b,:]+a1[b,:])) . w[h,:] + bias[h]
// OT h-tiles per wave share the A fragment. Grid: (H/(16*OT), B/16), block 32.
// ---------------------------------------------------------------------------
template <int F, int H, int OT>
__global__ __launch_bounds__(32) void lin1_wmma_kernel(
    const float* __restrict__ a0, const float* __restrict__ a1,
    const float* __restrict__ w,  const float* __restrict__ bias,
    float* __restrict__ out)
{
    const int h0   = blockIdx.x * 16 * OT;
    const int b0   = blockIdx.y * 16;
    const int lane = threadIdx.x;
    const int half = lane >> 4;
    const int rc   = lane & 15;

    const float* __restrict__ ar0 = a0 + (size_t)(b0 + rc) * F;
    const float* __restrict__ ar1 = a1 + (size_t)(b0 + rc) * F;
    const float* __restrict__ wr[OT];
#pragma unroll
    for (int ot = 0; ot < OT; ++ot) {
        wr[ot] = w + (size_t)(h0 + ot * 16 + rc) * F;
        __builtin_prefetch(wr[ot], 0, 1);
    }

    v8f acc[OT];
#pragma unroll
    for (int ot = 0; ot < OT; ++ot) acc[ot] = (v8f){};

    for (int kb = 0; kb < F; kb += 4) {
        const int k0 = kb + half * 2;
        const v2f x0 = *(const v2f*)(ar0 + k0);
        const v2f x1 = *(const v2f*)(ar1 + k0);
        v2f a;
        a.x = 0.5f * (x0.x + x1.x);
        a.y = 0.5f * (x0.y + x1.y);
#pragma unroll
        for (int ot = 0; ot < OT; ++ot) {
            const v2f b = *(const v2f*)(wr[ot] + k0);
            acc[ot] = __builtin_amdgcn_wmma_f32_16x16x4_f32(
                          false, a, false, b, (short)0, acc[ot], false, false);
        }
    }

#pragma unroll
    for (int ot = 0; ot < OT; ++ot) {
        const int h  = h0 + ot * 16 + rc;
        const float bv = bias[h];
#pragma unroll
        for (int j = 0; j < 8; ++j) {
            const int m = j + 8 * half;
            out[(size_t)(b0 + m) * H + h] = acc[ot][j] + bv;
        }
    }
}

// ---------------------------------------------------------------------------
// Head: lin2 (H->3), then log_softmax over cols 0..1, passthrough col 2.
// One block per batch row.
// ---------------------------------------------------------------------------
__global__ __launch_bounds__(256) void head_kernel(
    const float* __restrict__ x,    // (B, H) post-BN-relu
    const float* __restrict__ w,    // (3, H)
    const float* __restrict__ bias, // (3)
    float* __restrict__ out,        // (B, 3)
    int H)
{
    __shared__ float red[256];
    const int b = blockIdx.x;
    float dots[3];
#pragma unroll
    for (int j = 0; j < 3; ++j) {
        float s = 0.f;
        for (int k = threadIdx.x; k < H; k += 256)
            s += x[(size_t)b * H + k] * w[j * H + k];
        red[threadIdx.x] = s;
        __syncthreads();
        for (int st = 128; st > 0; st >>= 1) {
            if (threadIdx.x < st) red[threadIdx.x] += red[threadIdx.x + st];
            __syncthreads();
        }
        dots[j] = red[0] + bias[j];
        __syncthreads();
    }
    if (threadIdx.x == 0) {
        const float a = dots[0], c = dots[1];
        const float mx  = fmaxf(a, c);
        const float lse = mx + logf(expf(a - mx) + expf(c - mx));
        out[b * 3 + 0] = a - lse;
        out[b * 3 + 1] = c - lse;
        out[b * 3 + 2] = dots[2];
    }
}

// ---------------------------------------------------------------------------
// Host-side per-layer driver (templated so device shapes are compile-time).
// ---------------------------------------------------------------------------
template <int Cin, int Nin, int Cout, int Nout, int BStride>
static void run_layer(const float* inp, const int* ic, const int* ip,
                      const float* w, const float* bi,
                      const float* ga, const float* be,
                      float* convbuf, float* pdst,
                      float* scalebuf, float* shiftbuf, hipStream_t stream)
{
    constexpr int B = 256, OT = 4;
    dim3 gconv(Nin, Cout / (16 * OT), B / 16);
    conv_wmma_kernel<Cin, Nin, Cout, BStride, OT><<<gconv, 32, 0, stream>>>(
        inp, ic, w, bi, convbuf);

    const int ptotal = B * Cout * Nout;
    pool_kernel<<<(ptotal + 255) / 256, 256, 0, stream>>>(
        convbuf, ip, pdst, Nin, Nout, ptotal);

    bn_stats_kernel<<<Cout, 256, 0, stream>>>(
        pdst, B, Cout, Nout, ga, be, scalebuf, shiftbuf);

    bn_apply_relu_kernel<<<(ptotal + 255) / 256, 256, 0, stream>>>(
        pdst, pdst, scalebuf, shiftbuf, Cout, Nout, ptotal);
}

// ---------------------------------------------------------------------------
extern "C" void kernel_launch(void* const* d_in, const int* in_sizes, int n_in,
                              void* d_out, int out_size, void* d_ws, size_t ws_size,
                              hipStream_t stream)
{
    (void)in_sizes; (void)n_in; (void)out_size; (void)ws_size;

    const float* x = (const float*)d_in[0];
    const int*   idx_conv[4];
    const int*   idx_pool[4];
    const float *w[4], *bias[4], *gamma[4], *beta[4];
    for (int i = 0; i < 4; ++i) {
        idx_conv[i] = (const int*)  d_in[1 + 6 * i];
        idx_pool[i] = (const int*)  d_in[2 + 6 * i];
        w[i]        = (const float*)d_in[3 + 6 * i];
        bias[i]     = (const float*)d_in[4 + 6 * i];
        gamma[i]    = (const float*)d_in[5 + 6 * i];
        beta[i]     = (const float*)d_in[6 + 6 * i];
    }
    const float* lin1_w = (const float*)d_in[25];
    const float* lin1_b = (const float*)d_in[26];
    const float* gamma5 = (const float*)d_in[27];
    const float* beta5  = (const float*)d_in[28];
    const float* lin2_w = (const float*)d_in[29];
    const float* lin2_b = (const float*)d_in[30];

    // Workspace layout (floats).
    float* ws       = (float*)d_ws;
    float* convbuf  = ws;                              // 256*64*2048 = 33,554,432
    float* poolbuf  = convbuf + 33554432;              //  8,388,608
    float* act4_0   = poolbuf + 8388608;               //  1,048,576 (256*512*8)
    float* act4_1   = act4_0  + 1048576;               //  1,048,576
    float* lin1out  = act4_1  + 1048576;               //    524,288 (256*2048)
    float* scalebuf = lin1out + 524288;                //      2,048
    float* shiftbuf = scalebuf + 2048;                 //      2,048

    const int B = 256;

    for (int t = 0; t < 2; ++t) {
        const float* in0 = x + t * 2 * 2048;           // group slice of (B,4,2048)
        float* a4 = (t == 0) ? act4_0 : act4_1;

        // Layer 1: (B,2,2048) -> (B,64,512);  input b-stride 4*2048 (sliced).
        run_layer<2, 2048, 64, 512, 8192>(
            in0, idx_conv[0], idx_pool[0], w[0], bias[0], gamma[0], beta[0],
            convbuf, poolbuf, scalebuf, shiftbuf, stream);

        // Layer 2: (B,64,512) -> (B,128,128)
        run_layer<64, 512, 128, 128, 64 * 512>(
            poolbuf, idx_conv[1], idx_pool[1], w[1], bias[1], gamma[1], beta[1],
            convbuf, poolbuf, scalebuf, shiftbuf, stream);

        // Layer 3: (B,128,128) -> (B,256,32)
        run_layer<128, 128, 256, 32, 128 * 128>(
            poolbuf, idx_conv[2], idx_pool[2], w[2], bias[2], gamma[2], beta[2],
            convbuf, poolbuf, scalebuf, shiftbuf, stream);

        // Layer 4: (B,256,32) -> (B,512,8) -> persists in act4[t]
        run_layer<256, 32, 512, 8, 256 * 32>(
            poolbuf, idx_conv[3], idx_pool[3], w[3], bias[3], gamma[3], beta[3],
            convbuf, a4, scalebuf, shiftbuf, stream);
    }

    // Head: lin1 (fused group-mean) + BN + ReLU + lin2 + log_softmax.
    {
        constexpr int F = 4096, H = 2048, OT = 4;
        dim3 g(H / (16 * OT), B / 16);
        lin1_wmma_kernel<F, H, OT><<<g, 32, 0, stream>>>(
            act4_0, act4_1, lin1_w, lin1_b, lin1out);

        bn_stats_kernel<<<H, 256, 0, stream>>>(
            lin1out, B, H, 1, gamma5, beta5, scalebuf, shiftbuf);

        const int tt = B * H;
        bn_apply_relu_kernel<<<(tt + 255) / 256, 256, 0, stream>>>(
            lin1out, lin1out, scalebuf, shiftbuf, H, 1, tt);

        head_kernel<<<B, 256, 0, stream>>>(lin1out, lin2_w, lin2_b, (float*)d_out, H);
    }
}